// SparseSumOp_73710228734303
// MI455X (gfx1250) — compile-verified
//
#include <hip/hip_runtime.h>

typedef __attribute__((ext_vector_type(2))) float v2f;
typedef __attribute__((ext_vector_type(8))) float v8f;

#define BLOCK 256
#define GRID  1024
#define WAVES_PER_BLOCK (BLOCK / 32)
#define CHUNK 256   // floats per wave-superchunk: 2 x (32 lanes x float4)

// Stage 1: grid-stride over 256-float superchunks. Each wave issues two
// independent coalesced b128 loads (1 KB in flight), then folds the 256
// floats with four v_wmma_f32_16x16x4_f32 (B == ones) into four independent
// 16x16 fp32 accumulators.
__global__ __launch_bounds__(BLOCK)
void SparseSum_stage1(const float* __restrict__ vals, long long nchunks,
                      float* __restrict__ partials) {
    const int lane = threadIdx.x & 31;
    const int wib  = threadIdx.x >> 5;
    const long long w = (long long)blockIdx.x * WAVES_PER_BLOCK + wib;
    const long long W = (long long)gridDim.x * WAVES_PER_BLOCK;

    v8f acc0 = {};
    v8f acc1 = {};
    v8f acc2 = {};
    v8f acc3 = {};
    v2f ones;
    ones[0] = 1.0f;
    ones[1] = 1.0f;

    for (long long c = w; c < nchunks; c += W) {
        const float* base = vals + c * CHUNK + lane * 4;
        const float4 u = *reinterpret_cast<const float4*>(base);
        const float4 v = *reinterpret_cast<const float4*>(base + 128);
        v2f a0; a0[0] = u.x; a0[1] = u.y;
        v2f a1; a1[0] = u.z; a1[1] = u.w;
        v2f a2; a2[0] = v.x; a2[1] = v.y;
        v2f a3; a3[0] = v.z; a3[1] = v.w;
        // D = A x ones + C : every WMMA accumulates 64 loaded floats.
        acc0 = __builtin_amdgcn_wmma_f32_16x16x4_f32(
            false, a0, false, ones, (short)0, acc0, false, false);
        acc1 = __builtin_amdgcn_wmma_f32_16x16x4_f32(
            false, a1, false, ones, (short)0, acc1, false, false);
        acc2 = __builtin_amdgcn_wmma_f32_16x16x4_f32(
            false, a2, false, ones, (short)0, acc2, false, false);
        acc3 = __builtin_amdgcn_wmma_f32_16x16x4_f32(
            false, a3, false, ones, (short)0, acc3, false, false);
    }

    // Column 'n' of D replicates rowsum[m]; summing the 8 acc VGPRs per lane
    // yields sum(rows 0..7) in lanes 0-15 and sum(rows 8..15) in lanes 16-31.
    v8f acc = (acc0 + acc1) + (acc2 + acc3);
    float s = acc[0] + acc[1] + acc[2] + acc[3] +
              acc[4] + acc[5] + acc[6] + acc[7];
    float wave_total = __shfl(s, 0, 32) + __shfl(s, 16, 32);

    __shared__ float wsum[WAVES_PER_BLOCK];
    if (lane == 0) wsum[wib] = wave_total;
    __syncthreads();
    if (threadIdx.x == 0) {
        float b = 0.0f;
        for (int i = 0; i < WAVES_PER_BLOCK; ++i) b += wsum[i];
        partials[blockIdx.x] = b;   // every block writes, even if it did 0 chunks
    }
}

// Stage 2: one block reduces the block partials plus any tail elements
// (n % CHUNK leftovers) with a deterministic LDS tree.
__global__ __launch_bounds__(BLOCK)
void SparseSum_stage2(const float* __restrict__ partials, int nparts,
                      const float* __restrict__ vals, long long tail_start,
                      long long n, float* __restrict__ out) {
    __shared__ float smem[BLOCK];
    float s = 0.0f;
    for (int i = threadIdx.x; i < nparts; i += BLOCK) s += partials[i];
    for (long long i = tail_start + threadIdx.x; i < n; i += BLOCK)
        s += vals[i];
    smem[threadIdx.x] = s;
    __syncthreads();
    for (int off = BLOCK / 2; off > 0; off >>= 1) {
        if ((int)threadIdx.x < off) smem[threadIdx.x] += smem[threadIdx.x + off];
        __syncthreads();
    }
    if (threadIdx.x == 0) out[0] = smem[0];
}

extern "C" void kernel_launch(void* const* d_in, const int* in_sizes, int n_in,
                              void* d_out, int out_size, void* d_ws, size_t ws_size,
                              hipStream_t stream) {
    const float* vals = (const float*)d_in[0];   // values: fp32, NNZ elements
    // d_in[1] (indices) is unused: full reduction over all stored nonzeros.
    const long long n = (long long)in_sizes[0];
    const long long nchunks = n / CHUNK;
    const long long tail_start = nchunks * (long long)CHUNK;

    float* partials = (float*)d_ws;              // GRID floats of scratch

    SparseSum_stage1<<<GRID, BLOCK, 0, stream>>>(vals, nchunks, partials);
    SparseSum_stage2<<<1, BLOCK, 0, stream>>>(partials, GRID, vals, tail_start,
                                              n, (float*)d_out);
}